// DGL_GCN_15994458210646
// MI455X (gfx1250) — compile-verified
//
#include <hip/hip_runtime.h>
#include <hip/hip_bf16.h>

// ---- problem constants (match reference) ----
#define N_OBJ 4096
#define F_IN  256
#define HID   256
#define NCLS  128
#define NE    8
#define EPT   131072          // 2^17
#define EPT_LOG2 17

typedef float v2f __attribute__((ext_vector_type(2)));
typedef float v8f __attribute__((ext_vector_type(8)));

// ------------------------------------------------------------------
// zero fill
__global__ __launch_bounds__(256) void k_zero(float* __restrict__ p, int n) {
    int i = blockIdx.x * 256 + threadIdx.x;
    if (i < n) p[i] = 0.0f;
}

// ------------------------------------------------------------------
// per-etype degree count (float accumulation in L2)
__global__ __launch_bounds__(256) void k_count(const int* __restrict__ edges,
                                               float* __restrict__ cnt) {
    int gid = blockIdx.x * 256 + threadIdx.x;          // 0 .. NE*EPT-1
    int e = gid >> EPT_LOG2;
    int i = gid & (EPT - 1);
    int dst = edges[(size_t)e * 2 * EPT + EPT + i];
    atomicAdd(&cnt[e * N_OBJ + dst], 1.0f);
}

// cnt -> 1/max(cnt,1) in place
__global__ __launch_bounds__(256) void k_invdeg(float* __restrict__ cnt, int n) {
    int i = blockIdx.x * 256 + threadIdx.x;
    if (i < n) cnt[i] = 1.0f / fmaxf(cnt[i], 1.0f);
}

// ------------------------------------------------------------------
// Wh[e] = h @ W[e]^T + b[e]  using V_WMMA_F32_16X16X4_F32 (exact fp32).
// grid = (16, 4, NE), block = 256 (8 waves).
// Block tile: M=256 (8 waves x 32 rows), N=64. The 64 weight rows
// (B = W[e]^T columns) are staged in LDS once and shared by all 8 waves.
// Wave tile 32x64: 2 A fragments x 4 B fragments -> 8 WMMAs per k-step
// against 2 global b64 + 4 ds b64 loads.
#define BPAD 260   // 256 + 4 pad: (260 % 64) = 4 -> conflict-free lane access
__global__ __launch_bounds__(256) void k_gemm_etype(const float* __restrict__ h,
                                                    const float* __restrict__ W,
                                                    const float* __restrict__ bias,
                                                    float* __restrict__ Wh) {
    __shared__ float Bs[64 * BPAD];           // 66,560 B of the 320 KB WGP LDS
    const int lane = threadIdx.x & 31;
    const int wave = threadIdx.x >> 5;
    const int e    = blockIdx.z;
    const int K    = 256;
    const int n0   = blockIdx.y * 64;
    const int m0   = (blockIdx.x * 8 + wave) * 32;

    const float* We = W + (size_t)e * HID * K;

    // cooperative B tile load: W[e] rows n0..n0+63, full K
    for (int idx = threadIdx.x; idx < 64 * 64; idx += 256) {
        int row = idx >> 6;                   // 0..63
        int c4  = (idx & 63) << 2;            // 0..252 step 4
        float4 v = *(const float4*)(We + (size_t)(n0 + row) * K + c4);
        *(float4*)(&Bs[row * BPAD + c4]) = v;
    }
    __syncthreads();

    // WMMA f32 16x16x4 fragment addressing (ISA 7.12.2):
    //   A: lane l holds A[m0'+(l&15)][k + 2*(l>>4) + {0,1}]   (contiguous float2)
    //   B: lane l holds B[k + 2*(l>>4) + {0,1}][n0'+(l&15)] = W[n0'+(l&15)][k+...]
    const int khalf = (lane >> 4) * 2;
    const float* arow0 = h + (size_t)(m0 + (lane & 15)) * K + khalf;
    const float* arow1 = arow0 + (size_t)16 * K;
    const float* bl0 = &Bs[( 0 + (lane & 15)) * BPAD + khalf];
    const float* bl1 = &Bs[(16 + (lane & 15)) * BPAD + khalf];
    const float* bl2 = &Bs[(32 + (lane & 15)) * BPAD + khalf];
    const float* bl3 = &Bs[(48 + (lane & 15)) * BPAD + khalf];

    v8f c[8] = {};   // c[2*t+u]: N sub-tile t (0..3), M half u (0..1)
#pragma unroll 4
    for (int k = 0; k < K; k += 4) {
        v2f a0 = *(const v2f*)(arow0 + k);
        v2f a1 = *(const v2f*)(arow1 + k);
        v2f b0 = *(const v2f*)(bl0 + k);
        v2f b1 = *(const v2f*)(bl1 + k);
        v2f b2 = *(const v2f*)(bl2 + k);
        v2f b3 = *(const v2f*)(bl3 + k);
        c[0] = __builtin_amdgcn_wmma_f32_16x16x4_f32(false, a0, false, b0, (short)0, c[0], false, false);
        c[1] = __builtin_amdgcn_wmma_f32_16x16x4_f32(false, a1, false, b0, (short)0, c[1], false, false);
        c[2] = __builtin_amdgcn_wmma_f32_16x16x4_f32(false, a0, false, b1, (short)0, c[2], false, false);
        c[3] = __builtin_amdgcn_wmma_f32_16x16x4_f32(false, a1, false, b1, (short)0, c[3], false, false);
        c[4] = __builtin_amdgcn_wmma_f32_16x16x4_f32(false, a0, false, b2, (short)0, c[4], false, false);
        c[5] = __builtin_amdgcn_wmma_f32_16x16x4_f32(false, a1, false, b2, (short)0, c[5], false, false);
        c[6] = __builtin_amdgcn_wmma_f32_16x16x4_f32(false, a0, false, b3, (short)0, c[6], false, false);
        c[7] = __builtin_amdgcn_wmma_f32_16x16x4_f32(false, a1, false, b3, (short)0, c[7], false, false);
    }

    // C/D layout: VGPR r, lane l -> row base + r + 8*(l>>4), col n0' + (l&15)
    const int moff = 8 * (lane >> 4);
    float* out = Wh + (size_t)e * N_OBJ * HID;
#pragma unroll
    for (int t = 0; t < 4; ++t) {
        const int col = n0 + 16 * t + (lane & 15);
        const float bv = bias[e * HID + col];
#pragma unroll
        for (int u = 0; u < 2; ++u) {
            const int mb = m0 + 16 * u + moff;
            v8f cc = c[2 * t + u];
#pragma unroll
            for (int r = 0; r < 8; ++r)
                out[(size_t)(mb + r) * HID + col] = cc[r] + bv;
        }
    }
}

// ------------------------------------------------------------------
// scatter: agg[dst] += Wh[e][src] * invdeg[e][dst]  (one wave per edge)
__global__ __launch_bounds__(256) void k_scatter(const float* __restrict__ Wh,
                                                 const int* __restrict__ edges,
                                                 const float* __restrict__ invdeg,
                                                 float* __restrict__ agg) {
    const int lane = threadIdx.x & 31;
    const int wave = threadIdx.x >> 5;
    const long long eidx = (long long)blockIdx.x * 8 + wave;   // 0 .. NE*EPT-1
    const int e = (int)(eidx >> EPT_LOG2);
    const int i = (int)(eidx & (EPT - 1));
    const int* ed = edges + (size_t)e * 2 * EPT;
    const int src = ed[i];
    const int dst = ed[EPT + i];
    const float s = invdeg[e * N_OBJ + dst];
    const float4* msg = (const float4*)(Wh + ((size_t)e * N_OBJ + src) * HID);
    float* dout = agg + (size_t)dst * HID;
#pragma unroll
    for (int t = 0; t < 2; ++t) {
        float4 v = msg[lane + 32 * t];
        int base = (lane + 32 * t) * 4;
        atomicAdd(dout + base + 0, v.x * s);
        atomicAdd(dout + base + 1, v.y * s);
        atomicAdd(dout + base + 2, v.z * s);
        atomicAdd(dout + base + 3, v.w * s);
    }
}

// in-place relu
__global__ __launch_bounds__(256) void k_relu(float* __restrict__ p, int n) {
    int i = blockIdx.x * 256 + threadIdx.x;
    if (i < n) p[i] = fmaxf(p[i], 0.0f);
}

// ------------------------------------------------------------------
// fc1: z[r] = relu(fc1_w[r,:] . flat + fc1_b[r]); one block per output row.
// Streams the 1 GiB weight matrix at HBM rate with float4 loads.
__global__ __launch_bounds__(256) void k_fc1(const float* __restrict__ w,
                                             const float* __restrict__ bias,
                                             const float* __restrict__ flat,
                                             float* __restrict__ z) {
    __shared__ float red[256];
    const int r = blockIdx.x;
    const float4* wr = (const float4*)(w + (size_t)r * (N_OBJ * HID));
    const float4* fr = (const float4*)flat;
    float acc = 0.0f;
    for (int j = threadIdx.x; j < (N_OBJ * HID) / 4; j += 256) {
        float4 a = wr[j];
        float4 x = fr[j];
        acc += a.x * x.x + a.y * x.y + a.z * x.z + a.w * x.w;
    }
    red[threadIdx.x] = acc;
    __syncthreads();
    for (int s = 128; s > 0; s >>= 1) {
        if (threadIdx.x < s) red[threadIdx.x] += red[threadIdx.x + s];
        __syncthreads();
    }
    if (threadIdx.x == 0) z[r] = fmaxf(red[0] + bias[r], 0.0f);
}

// fc2 + sigmoid (tiny)
__global__ __launch_bounds__(128) void k_fc2(const float* __restrict__ w,
                                             const float* __restrict__ bias,
                                             const float* __restrict__ z,
                                             float* __restrict__ out) {
    const int c = threadIdx.x;
    float acc = bias[c];
#pragma unroll 8
    for (int k = 0; k < HID; ++k) acc += w[c * HID + k] * z[k];
    out[c] = 1.0f / (1.0f + __expf(-acc));
}

// ------------------------------------------------------------------
extern "C" void kernel_launch(void* const* d_in, const int* in_sizes, int n_in,
                              void* d_out, int out_size, void* d_ws, size_t ws_size,
                              hipStream_t stream) {
    const float* feat   = (const float*)d_in[0];
    const int*   edges  = (const int*)d_in[1];     // harness delivers integer inputs as int32
    const float* W[3]   = {(const float*)d_in[2], (const float*)d_in[4], (const float*)d_in[6]};
    const float* B[3]   = {(const float*)d_in[3], (const float*)d_in[5], (const float*)d_in[7]};
    const float* fc1_w  = (const float*)d_in[8];
    const float* fc1_b  = (const float*)d_in[9];
    const float* fc2_w  = (const float*)d_in[10];
    const float* fc2_b  = (const float*)d_in[11];
    float* out = (float*)d_out;

    // workspace layout (floats)
    char* ws = (char*)d_ws;
    float* Wh     = (float*)ws;                                   // NE*N*H   = 32 MB
    float* hA     = (float*)(ws + (size_t)NE * N_OBJ * HID * 4);  // N*H      = 4 MB
    float* hB     = hA + (size_t)N_OBJ * HID;                     // N*H      = 4 MB
    float* invdeg = hB + (size_t)N_OBJ * HID;                     // NE*N     = 128 KB
    float* z      = invdeg + (size_t)NE * N_OBJ;                  // H floats

    const int nH   = N_OBJ * HID;          // 1,048,576
    const int nDeg = NE * N_OBJ;           // 32,768

    // degrees (edges are constant across layers -> compute once per call)
    k_zero<<<(nDeg + 255) / 256, 256, 0, stream>>>(invdeg, nDeg);
    k_count<<<(NE * EPT) / 256, 256, 0, stream>>>(edges, invdeg);
    k_invdeg<<<(nDeg + 255) / 256, 256, 0, stream>>>(invdeg, nDeg);

    const float* h_in = feat;
    float* h_out = hA;
    for (int l = 0; l < 3; ++l) {
        k_gemm_etype<<<dim3(16, 4, NE), 256, 0, stream>>>(h_in, W[l], B[l], Wh);
        k_zero<<<nH / 256, 256, 0, stream>>>(h_out, nH);
        k_scatter<<<(NE * EPT) / 8, 256, 0, stream>>>(Wh, edges, invdeg, h_out);
        k_relu<<<nH / 256, 256, 0, stream>>>(h_out, nH);
        h_in  = h_out;
        h_out = (h_out == hA) ? hB : hA;
    }
    // h_in now points at the final node features

    k_fc1<<<HID, 256, 0, stream>>>(fc1_w, fc1_b, h_in, z);
    k_fc2<<<1, 128, 0, stream>>>(fc2_w, fc2_b, z, out);
}